// DH_FK_60627758350504
// MI455X (gfx1250) — compile-verified
//
#include <hip/hip_runtime.h>
#include <stdint.h>

// ---------------------------------------------------------------------------
// DH forward kinematics, B=1M elements. Memory-bound (~36 B/elem, ~1.6 us HBM
// floor). Strategy: TDM (tensor_load_to_lds / tensor_store_from_lds) for bulk
// contiguous staging of the (256x6) input tile and (256x3) output tile per
// block, rigid-transform (R,p) chain composition in fp32 VALU (~180 FMA +
// 6 sincos per element), per-block precompute of alpha trig.
// ---------------------------------------------------------------------------

typedef unsigned int v4u __attribute__((ext_vector_type(4)));
typedef int          v4i __attribute__((ext_vector_type(4)));
typedef int          v8i __attribute__((ext_vector_type(8)));

#if defined(__HIP_DEVICE_COMPILE__) &&                                   \
    __has_builtin(__builtin_amdgcn_tensor_load_to_lds) &&                \
    __has_builtin(__builtin_amdgcn_tensor_store_from_lds) &&             \
    __has_builtin(__builtin_amdgcn_s_wait_tensorcnt)
#define USE_TDM 1
#else
#define USE_TDM 0
#endif

__device__ __forceinline__ uint32_t lds_off(const void* p) {
  // low 32 bits of a flat pointer into the LDS aperture == LDS byte offset
  return (uint32_t)(uintptr_t)p;
}

#if USE_TDM
// Build a D# for a 1-D contiguous copy of `ndw` dwords (ndw <= 65535).
// Group0: count=1 | lds_addr | global_addr[56:0] | type=2.
// Group1: data_size=4B, tensor_dim0=ndw, tensor_dim1=1, tile_dim0=ndw,
//         tile_dim1/2=0 (unused), dim0_stride=ndw. Groups 2/3 zero (<=2D).
__device__ __forceinline__ void tdm_desc_1d(uint64_t ga, uint32_t lds,
                                            uint32_t ndw, v4u& g0, v8i& g1) {
  g0.x = 1u;                                   // count=1, user mode, no gather
  g0.y = lds;                                  // LDS byte address
  g0.z = (uint32_t)ga;                         // global_addr[31:0]
  g0.w = ((uint32_t)(ga >> 32) & 0x01FFFFFFu)  // global_addr[56:32]
         | 0x80000000u;                        // type=2 ("image") in [127:126]
  g1[0] = (int)(2u << 16);                     // wg_mask=0, data_size=2 (4B)
  g1[1] = (int)((ndw & 0xFFFFu) << 16);        // tensor_dim0[15:0]
  g1[2] = (int)((ndw >> 16) | (1u << 16));     // tensor_dim0[31:16], dim1=1
  g1[3] = (int)((ndw & 0xFFFFu) << 16);        // dim1_hi=0, tile_dim0=ndw
  g1[4] = 0;                                   // tile_dim1=0, tile_dim2=0
  g1[5] = (int)ndw;                            // tensor_dim0_stride lo32
  g1[6] = 0;
  g1[7] = 0;
}

__device__ __forceinline__ void tdm_load_1d(const void* g, uint32_t lds,
                                            uint32_t ndw) {
  v4u g0; v8i g1;
  tdm_desc_1d((uint64_t)g, lds, ndw, g0, g1);
  v4i z4 = {0, 0, 0, 0};
  v8i z8 = {0, 0, 0, 0, 0, 0, 0, 0};
  // 6-arg form (amdgpu-toolchain / clang-23): groups 0..3 + zero v8i + cpol
  __builtin_amdgcn_tensor_load_to_lds(g0, g1, z4, z4, z8, 0);
}

__device__ __forceinline__ void tdm_store_1d(void* g, uint32_t lds,
                                             uint32_t ndw) {
  v4u g0; v8i g1;
  tdm_desc_1d((uint64_t)g, lds, ndw, g0, g1);
  v4i z4 = {0, 0, 0, 0};
  v8i z8 = {0, 0, 0, 0, 0, 0, 0, 0};
  __builtin_amdgcn_tensor_store_from_lds(g0, g1, z4, z4, z8, 0);
}
#endif  // USE_TDM

// ---------------------------------------------------------------------------
// Main kernel: one thread = one batch element, 256 threads (8 waves) / block.
// ---------------------------------------------------------------------------
__global__ __launch_bounds__(256) void fk_main(const float* __restrict__ q,
                                               const float* __restrict__ dh,
                                               float* __restrict__ out) {
  __shared__ float sQ[256 * 6];   // staged joint angles (contiguous tile)
  __shared__ float sP[256 * 3];   // staged positions (contiguous tile)
  __shared__ float prm[6][5];     // per-joint: th0, d, a, sin(alpha), cos(alpha)

  const uint32_t tid  = threadIdx.x;
  const size_t   base = (size_t)blockIdx.x * 256u;

#if USE_TDM
  if (tid < 32u)                          // wave 0 issues the DMA
    tdm_load_1d(q + base * 6u, lds_off(sQ), 256u * 6u);
#else
  for (uint32_t k = tid; k < 256u * 6u; k += 256u) sQ[k] = q[base * 6u + k];
#endif

  if (tid < 6u) {                         // uniform per-block DH precompute
    const float th0 = dh[tid * 4 + 0];
    const float d   = dh[tid * 4 + 1];
    const float a   = dh[tid * 4 + 2];
    const float al  = dh[tid * 4 + 3];
    float sa, ca;
    __sincosf(al, &sa, &ca);
    prm[tid][0] = th0; prm[tid][1] = d; prm[tid][2] = a;
    prm[tid][3] = sa;  prm[tid][4] = ca;
  }
#if USE_TDM
  if (tid < 32u) __builtin_amdgcn_s_wait_tensorcnt(0);
#endif
  __syncthreads();

  // Rigid-transform chain: T' = T * L_j with L_j the DH link transform.
  // Keep R (3x3) and p (3). Columns of R: c0=(r00,r10,r20) etc.
  float r00, r01, r02, r10, r11, r12, r20, r21, r22, px, py, pz;
  {
    float st, ct;
    __sincosf(sQ[tid * 6 + 0] + prm[0][0], &st, &ct);
    const float d = prm[0][1], a = prm[0][2], sa = prm[0][3], ca = prm[0][4];
    r00 = ct;  r01 = -st * ca; r02 =  st * sa; px = a * ct;
    r10 = st;  r11 =  ct * ca; r12 = -ct * sa; py = a * st;
    r20 = 0.f; r21 =  sa;      r22 =  ca;      pz = d;
  }
#pragma unroll
  for (int j = 1; j < 6; ++j) {
    float st, ct;
    __sincosf(sQ[tid * 6 + j] + prm[j][0], &st, &ct);
    const float d = prm[j][1], a = prm[j][2], sa = prm[j][3], ca = prm[j][4];
    // u = R*(ct,st,0),  v = R*(-st,ct,0)
    const float ux = fmaf(ct, r00,  st * r01);
    const float uy = fmaf(ct, r10,  st * r11);
    const float uz = fmaf(ct, r20,  st * r21);
    const float vx = fmaf(ct, r01, -st * r00);
    const float vy = fmaf(ct, r11, -st * r10);
    const float vz = fmaf(ct, r21, -st * r20);
    // new c1 = ca*v + sa*c2 ; new c2 = -sa*v + ca*c2 ; p += a*u + d*c2
    const float w1x = fmaf(ca, vx,  sa * r02);
    const float w1y = fmaf(ca, vy,  sa * r12);
    const float w1z = fmaf(ca, vz,  sa * r22);
    const float w2x = fmaf(ca, r02, -sa * vx);
    const float w2y = fmaf(ca, r12, -sa * vy);
    const float w2z = fmaf(ca, r22, -sa * vz);
    px = fmaf(a, ux, fmaf(d, r02, px));
    py = fmaf(a, uy, fmaf(d, r12, py));
    pz = fmaf(a, uz, fmaf(d, r22, pz));
    r00 = ux;  r10 = uy;  r20 = uz;
    r01 = w1x; r11 = w1y; r21 = w1z;
    r02 = w2x; r12 = w2y; r22 = w2z;
  }

  sP[tid * 3 + 0] = px;
  sP[tid * 3 + 1] = py;
  sP[tid * 3 + 2] = pz;
  __syncthreads();

#if USE_TDM
  if (tid < 32u) {
    tdm_store_1d(out + base * 3u, lds_off(sP), 256u * 3u);
    __builtin_amdgcn_s_wait_tensorcnt(0);  // keep wave (and LDS) alive until done
  }
#else
  for (uint32_t k = tid; k < 256u * 3u; k += 256u) out[base * 3u + k] = sP[k];
#endif
}

// ---------------------------------------------------------------------------
// Tail kernel (generic safety; never launched for B = 1<<20 since 256 | B).
// ---------------------------------------------------------------------------
__global__ void fk_tail(const float* __restrict__ q,
                        const float* __restrict__ dh,
                        float* __restrict__ out, int start, int n) {
  const int i = start + (int)(blockIdx.x * blockDim.x + threadIdx.x);
  if (i >= n) return;
  float r00, r01, r02, r10, r11, r12, r20, r21, r22, px, py, pz;
  {
    float sa, ca; __sincosf(dh[3], &sa, &ca);
    float st, ct; __sincosf(q[(size_t)i * 6 + 0] + dh[0], &st, &ct);
    const float d = dh[1], a = dh[2];
    r00 = ct;  r01 = -st * ca; r02 =  st * sa; px = a * ct;
    r10 = st;  r11 =  ct * ca; r12 = -ct * sa; py = a * st;
    r20 = 0.f; r21 =  sa;      r22 =  ca;      pz = d;
  }
  for (int j = 1; j < 6; ++j) {
    float sa, ca; __sincosf(dh[j * 4 + 3], &sa, &ca);
    float st, ct; __sincosf(q[(size_t)i * 6 + j] + dh[j * 4 + 0], &st, &ct);
    const float d = dh[j * 4 + 1], a = dh[j * 4 + 2];
    const float ux = fmaf(ct, r00,  st * r01), uy = fmaf(ct, r10,  st * r11), uz = fmaf(ct, r20,  st * r21);
    const float vx = fmaf(ct, r01, -st * r00), vy = fmaf(ct, r11, -st * r10), vz = fmaf(ct, r21, -st * r20);
    const float w1x = fmaf(ca, vx,  sa * r02), w1y = fmaf(ca, vy,  sa * r12), w1z = fmaf(ca, vz,  sa * r22);
    const float w2x = fmaf(ca, r02, -sa * vx), w2y = fmaf(ca, r12, -sa * vy), w2z = fmaf(ca, r22, -sa * vz);
    px = fmaf(a, ux, fmaf(d, r02, px));
    py = fmaf(a, uy, fmaf(d, r12, py));
    pz = fmaf(a, uz, fmaf(d, r22, pz));
    r00 = ux;  r10 = uy;  r20 = uz;
    r01 = w1x; r11 = w1y; r21 = w1z;
    r02 = w2x; r12 = w2y; r22 = w2z;
  }
  out[(size_t)i * 3 + 0] = px;
  out[(size_t)i * 3 + 1] = py;
  out[(size_t)i * 3 + 2] = pz;
}

extern "C" void kernel_launch(void* const* d_in, const int* in_sizes, int n_in,
                              void* d_out, int out_size, void* d_ws, size_t ws_size,
                              hipStream_t stream) {
  const float* q  = (const float*)d_in[0];   // (B, 6) f32
  const float* dh = (const float*)d_in[1];   // (6, 4) f32
  float* out = (float*)d_out;                // (B, 3) f32
  const int B    = in_sizes[0] / 6;
  const int full = B / 256;                  // full 256-element tiles
  if (full > 0)
    fk_main<<<full, 256, 0, stream>>>(q, dh, out);
  const int rem = B - full * 256;
  if (rem > 0)
    fk_tail<<<(rem + 255) / 256, 256, 0, stream>>>(q, dh, out, full * 256, B);
}